// PoseOptimizerLayer_44143673868793
// MI455X (gfx1250) — compile-verified
//
#include <hip/hip_runtime.h>
#include <hip/hip_bf16.h>

// ---------------------------------------------------------------------------
// Q = build_q_matrix(associations, pt_a, pt_b)
// Everything reduces to S = F^T * A * G  (4x4 per batch) with
//   F[i] = [1, x_a, y_a, x_a^2+y_a^2],  G[j] = [1, x_b, y_b, x_b^2+y_b^2]
// Stage 1: WMMA f32 16x16x4 streams A (HBM-bound: 256 MB @ 23.3 TB/s ~ 11 us),
//          accumulating H = A_block * G per 16-row block; tiny VALU stage
//          folds F^T * H -> 4x4 partial per (batch, i-block, j-segment).
// Stage 2: deterministic reduction of partials + Q assembly (25 floats/batch).
// ---------------------------------------------------------------------------

typedef float v2f __attribute__((ext_vector_type(2)));
typedef float v8f __attribute__((ext_vector_type(8)));

#define NB_MAX   2048            // reference NUM_B
#define GSTRIDE  (NB_MAX + 16)   // row stride (floats); +16 staggers LDS banks
#define JSPLIT   4               // j-segments per i-block
#define WAVES_PB 4               // waves per block (128 threads, wave32)

__global__ __launch_bounds__(128) void qmat_stage1(
    const float* __restrict__ A,     // [B][Na][Nb]
    const float* __restrict__ pa,    // [B][Na][2]
    const float* __restrict__ pb,    // [B][Nb][2]
    float* __restrict__ partial,     // [B][iblocks][JSPLIT][16]
    int Na, int Nb, int iblocks)
{
    // Transposed G table + one all-zero row (row 4) so every lane does an
    // unconditional ds_load_b64 for its B-operand pair (no exec masking).
    __shared__ float Gt[5 * GSTRIDE];
    __shared__ float Hs[WAVES_PB][16 * 16];   // per-wave D spill for stage-2

    const int lane = threadIdx.x & 31;
    const int wave = threadIdx.x >> 5;
    const int blocksPerBatch = (iblocks * JSPLIT) / WAVES_PB;
    const int b    = blockIdx.x / blocksPerBatch;
    const int task = (blockIdx.x % blocksPerBatch) * WAVES_PB + wave;
    const int iblk = task / JSPLIT;
    const int jseg = task % JSPLIT;

    // ---- build G table cooperatively (one batch per block) ----
    const float* pbb = pb + (size_t)b * Nb * 2;
    for (int j = threadIdx.x; j < Nb; j += blockDim.x) {
        float x = pbb[2 * j + 0];
        float y = pbb[2 * j + 1];
        Gt[0 * GSTRIDE + j] = 1.0f;
        Gt[1 * GSTRIDE + j] = x;
        Gt[2 * GSTRIDE + j] = y;
        Gt[3 * GSTRIDE + j] = x * x + y * y;
        Gt[4 * GSTRIDE + j] = 0.0f;           // zero row for N >= 4 lanes
    }
    __syncthreads();

    // ---- hot loop: D += A_tile(16x4) * B_tile(4x16) via v_wmma_f32_16x16x4_f32
    // A operand layout: lanes 0-15: M=lane, K={0,1}; lanes 16-31: M=lane-16, K={2,3}
    // B operand layout: lanes 0-15: N=lane, K={0,1}; lanes 16-31: N=lane-16, K={2,3}
    const int half = lane >> 4;            // which K-pair this lane supplies
    const int m    = lane & 15;            // M for A-operand, N for B-operand
    const int row  = iblk * 16 + m;
    const size_t rowbase = ((size_t)b * Na + row) * Nb;
    const int jchunk = Nb / JSPLIT;
    int col = jseg * jchunk + 2 * half;    // this lane's first column of the K-slab

    const int grow = (m < 4) ? m : 4;      // N>=4 lanes read the zero row
    const float* gptr = &Gt[grow * GSTRIDE];

    v8f acc = {};
    const int steps = jchunk / 4;
#pragma unroll 4
    for (int t = 0; t < steps; ++t) {
        __builtin_prefetch(A + rowbase + col + 256, 0, 3);   // ~1KB ahead on row
        v2f av = *(const v2f*)(A + rowbase + col);           // A[row][col..col+1]
        v2f bv = *(const v2f*)(gptr + col);                  // g(col), g(col+1)
        acc = __builtin_amdgcn_wmma_f32_16x16x4_f32(
                  false, av, false, bv, (short)0, acc, false, false);
        col += 4;
    }

    // ---- spill D (H = A_block * G) to LDS: C/D layout M = r + 8*half, N = m
    float* hs = &Hs[wave][0];
#pragma unroll
    for (int r = 0; r < 8; ++r) {
        hs[(r + 8 * half) * 16 + m] = acc[r];
    }
    __syncthreads();   // all waves run identical trip counts; also orders DS ops

    // ---- stage-2: S[p][q] = sum_i F[i0+i][p] * H[i][q]  (16 lanes, 256 MACs)
    if (lane < 16) {
        const int p = lane >> 2;
        const int q = lane & 3;
        const float* paa = pa + ((size_t)b * Na + iblk * 16) * 2;
        float s = 0.0f;
#pragma unroll
        for (int i = 0; i < 16; ++i) {
            float x = paa[2 * i + 0];
            float y = paa[2 * i + 1];
            float f = (p == 0) ? 1.0f : (p == 1) ? x : (p == 2) ? y : (x * x + y * y);
            s += f * hs[i * 16 + q];
        }
        partial[(((size_t)b * iblocks + iblk) * JSPLIT + jseg) * 16 + lane] = s;
    }
}

__global__ __launch_bounds__(32) void qmat_stage2(
    const float* __restrict__ partial, float* __restrict__ out, int nseg)
{
    const int b = blockIdx.x;
    const int t = threadIdx.x;
    __shared__ float S[16];

    if (t < 16) {
        const float* base = partial + (size_t)b * nseg * 16 + t;
        float s = 0.0f;
        for (int seg = 0; seg < nseg; ++seg) s += base[(size_t)seg * 16];
        S[t] = s;
    }
    __syncthreads();

    if (t == 0) {
        // S[p*4+q] = sum_ij A_ij f_p(i) g_q(j)
        float w     = S[0];
        float q01   = -S[1];          // -sum w_b x_b
        float q02   = -S[2];          // -sum w_b y_b
        float q00   =  S[3];          //  sum w_b (x_b^2+y_b^2)
        float q13   =  S[4];          //  sum w_a x_a
        float xa_xb =  S[5];
        float xa_yb =  S[6];
        float q23   =  S[8];          //  sum w_a y_a
        float ya_xb =  S[9];
        float ya_yb =  S[10];
        float q33   =  S[12];         //  sum w_a (x_a^2+y_a^2)
        float q14   = -q23;
        float q24   =  q13;
        float q03   = -(xa_xb + ya_yb);
        float q04   =  ya_xb - xa_yb;

        float* Q = out + (size_t)b * 25;
        Q[0]  = q00; Q[1]  = q01; Q[2]  = q02; Q[3]  = q03; Q[4]  = q04;
        Q[5]  = q01; Q[6]  = w;   Q[7]  = 0.f; Q[8]  = q13; Q[9]  = q14;
        Q[10] = q02; Q[11] = 0.f; Q[12] = w;   Q[13] = q23; Q[14] = q24;
        Q[15] = q03; Q[16] = q13; Q[17] = q23; Q[18] = q33; Q[19] = 0.f;
        Q[20] = q04; Q[21] = q14; Q[22] = q24; Q[23] = 0.f; Q[24] = q33;
    }
}

extern "C" void kernel_launch(void* const* d_in, const int* in_sizes, int n_in,
                              void* d_out, int out_size, void* d_ws, size_t ws_size,
                              hipStream_t stream)
{
    (void)n_in; (void)ws_size;
    const float* A  = (const float*)d_in[0];   // associations [B][Na][Nb]
    const float* pa = (const float*)d_in[1];   // pt_in_a      [B][Na][2]
    const float* pb = (const float*)d_in[2];   // pt_in_b      [B][Nb][2]
    float* out = (float*)d_out;                // Q            [B][5][5]

    const int B  = out_size / 25;
    const int Na = in_sizes[1] / (2 * B);
    const int Nb = in_sizes[2] / (2 * B);
    const int iblocks = Na / 16;

    float* partial = (float*)d_ws;             // B*iblocks*JSPLIT*16 floats (512 KB)

    const int blocksPerBatch = (iblocks * JSPLIT) / WAVES_PB;
    qmat_stage1<<<dim3(B * blocksPerBatch), 128, 0, stream>>>(
        A, pa, pb, partial, Na, Nb, iblocks);
    qmat_stage2<<<dim3(B), 32, 0, stream>>>(partial, out, iblocks * JSPLIT);
}